// CompGraphConv_37263136260548
// MI455X (gfx1250) — compile-verified
//
#include <hip/hip_runtime.h>

typedef float v2f __attribute__((ext_vector_type(2)));
typedef float v8f __attribute__((ext_vector_type(8)));

#define DD 64      // feature dim (in_dim == out_dim == 64)
#define SPITCH 68  // LDS row pitch (floats): 64 + 4 pad -> conflict-free fragment reads

// ---------------------------------------------------------------------------
// fp32 WMMA: D[16x16] = A[16x4] * B[4x16] + C   (exact fp32 path on CDNA5)
// A layout: lane l<16: M=l, VGPR0=K0,VGPR1=K1 ; lane>=16: M=l-16, K2/K3
// B layout: lane l<16: N=l, VGPR0=K0,VGPR1=K1 ; lane>=16: N=l-16, K2/K3
// C/D layout: VGPR i: lanes0-15 -> (M=i, N=lane); lanes16-31 -> (M=i+8, N=lane-16)
// ---------------------------------------------------------------------------
static __device__ __forceinline__ v8f wmma_f32x4(v2f a, v2f b, v8f c) {
  return __builtin_amdgcn_wmma_f32_16x16x4_f32(
      /*neg_a=*/false, a, /*neg_b=*/false, b,
      /*c_mod=*/(short)0, c, /*reuse_a=*/false, /*reuse_b=*/false);
}

// ---------------------------------------------------------------------------
// Zero workspace accumulators
// ---------------------------------------------------------------------------
__global__ void zero_kernel(float* __restrict__ p, long long n) {
  long long i = (long long)blockIdx.x * blockDim.x + threadIdx.x;
  long long stride = (long long)gridDim.x * blockDim.x;
  for (; i < n; i += stride) p[i] = 0.0f;
}

// ---------------------------------------------------------------------------
// Edge phase: one wave32 per edge. comp = n_feats[src] - r_feats[etype]
// scattered (atomicAdd f32) into accI or accO at row dst; lane0 bumps count.
// Each lane handles 2 contiguous floats -> fully coalesced 256B row traffic.
// All hot state (acc: 26MB, n_feats: 12.8MB) is L2-resident on MI455X (192MB).
// ---------------------------------------------------------------------------
__global__ void edge_scatter_kernel(const float* __restrict__ n_feats,
                                    const float* __restrict__ r_feats,
                                    const int* __restrict__ src,
                                    const int* __restrict__ dst,
                                    const int* __restrict__ etype,
                                    const int* __restrict__ num_rels_p,
                                    float* __restrict__ accI,
                                    float* __restrict__ accO,
                                    float* __restrict__ cntI,
                                    float* __restrict__ cntO,
                                    int E) {
  int wave = (int)((blockIdx.x * blockDim.x + threadIdx.x) >> 5);
  int lane = threadIdx.x & 31;
  if (wave >= E) return;

  int s = src[wave];
  int d = dst[wave];
  int t = etype[wave];
  int half = num_rels_p[0] >> 1;
  bool isI = (t < half);

  const float2* xs = (const float2*)(n_feats + (size_t)s * DD);
  const float2* xr = (const float2*)(r_feats + (size_t)t * DD);
  float2 a = xs[lane];
  float2 b = xr[lane];
  float cx = a.x - b.x;
  float cy = a.y - b.y;

  float* acc = isI ? accI : accO;
  size_t base = (size_t)d * DD + 2 * lane;
  atomicAdd(&acc[base + 0], cx);
  atomicAdd(&acc[base + 1], cy);
  if (lane == 0) atomicAdd(isI ? &cntI[d] : &cntO[d], 1.0f);
}

// ---------------------------------------------------------------------------
// Node phase: 128 threads = 4 waves; W_I/W_O staged once into LDS and shared
// by the 4 waves (cuts weight L2 traffic 4x; ds_load latency << vmem).
// One wave32 per 16-node tile:
//   out[16,64] = (accI@W_I^T + accO@W_O^T + cntI*b_I + cntO*b_O) / max(deg,1)
// I- and O-path matmuls chain into the same WMMA accumulator.
// Per tile: 16 K-steps x 4 column tiles x 2 paths = 128 v_wmma_f32_16x16x4_f32
// ---------------------------------------------------------------------------
__global__ void __launch_bounds__(128)
node_transform_kernel(const float* __restrict__ accI,
                      const float* __restrict__ accO,
                      const float* __restrict__ cntI,
                      const float* __restrict__ cntO,
                      const float* __restrict__ W_I_w,
                      const float* __restrict__ W_I_b,
                      const float* __restrict__ W_O_w,
                      const float* __restrict__ W_O_b,
                      float* __restrict__ out,
                      int numTiles) {
  __shared__ float sWI[DD * SPITCH];
  __shared__ float sWO[DD * SPITCH];

  // Cooperative staging: 64x64 f32 each, float4 rows (272B pitch keeps 16B align)
  {
    const float4* gI = (const float4*)W_I_w;
    const float4* gO = (const float4*)W_O_w;
    int tid = threadIdx.x;
#pragma unroll
    for (int i = 0; i < (DD * DD / 4) / 128; ++i) {  // 8 iters
      int idx = tid + i * 128;        // float4 index into 64x64
      int r = idx >> 4;               // row
      int c4 = idx & 15;              // float4 column
      *(float4*)(&sWI[r * SPITCH + c4 * 4]) = gI[idx];
      *(float4*)(&sWO[r * SPITCH + c4 * 4]) = gO[idx];
    }
  }
  __syncthreads();

  int tile = blockIdx.x * 4 + (threadIdx.x >> 5);
  if (tile >= numTiles) return;  // wave-uniform: full EXEC at every WMMA

  int lane = threadIdx.x & 31;
  int lm   = lane & 15;          // fragment row (A) / column (B)
  int hs   = lane >> 4;          // half-wave select
  int kb   = hs * 2;             // K sub-offset inside a K=4 step
  int mOff = hs * 8;             // C/D row offset for this half-wave
  int rowBase = tile * 16;

  // Per-lane edge counts + mean denominators for the 8 C/D rows this lane owns
  float cI8[8], cO8[8], dg8[8];
#pragma unroll
  for (int i = 0; i < 8; ++i) {
    int m = rowBase + i + mOff;
    cI8[i] = cntI[m];
    cO8[i] = cntO[m];
    dg8[i] = fmaxf(cI8[i] + cO8[i], 1.0f);
  }

  const float* aIrow = accI + (size_t)(rowBase + lm) * DD;
  const float* aOrow = accO + (size_t)(rowBase + lm) * DD;
  // B fragment rows in LDS: B[k][n] = W[n][k]; column n = t*16 + lm
  const float* sIrow[4];
  const float* sOrow[4];
#pragma unroll
  for (int t = 0; t < 4; ++t) {
    sIrow[t] = sWI + (t * 16 + lm) * SPITCH;
    sOrow[t] = sWO + (t * 16 + lm) * SPITCH;
  }

  v8f c[4];
#pragma unroll
  for (int t = 0; t < 4; ++t) c[t] = (v8f){0.f, 0.f, 0.f, 0.f, 0.f, 0.f, 0.f, 0.f};

#pragma unroll
  for (int k0 = 0; k0 < DD; k0 += 4) {
    v2f aI = *(const v2f*)(aIrow + k0 + kb);   // streaming global (L2-resident)
    v2f aO = *(const v2f*)(aOrow + k0 + kb);
#pragma unroll
    for (int t = 0; t < 4; ++t) {
      v2f bI = *(const v2f*)(sIrow[t] + k0 + kb);  // ds_load_b64, conflict-free
      v2f bO = *(const v2f*)(sOrow[t] + k0 + kb);
      c[t] = wmma_f32x4(aI, bI, c[t]);
      c[t] = wmma_f32x4(aO, bO, c[t]);
    }
  }

  // Epilogue: + cntI*b_I + cntO*b_O, divide by degree, store
#pragma unroll
  for (int t = 0; t < 4; ++t) {
    int n = t * 16 + lm;
    float bI = W_I_b[n];
    float bO = W_O_b[n];
#pragma unroll
    for (int i = 0; i < 8; ++i) {
      int m = rowBase + i + mOff;
      float v = (c[t][i] + cI8[i] * bI + cO8[i] * bO) / dg8[i];
      out[(size_t)m * DD + n] = v;
    }
  }
}

// ---------------------------------------------------------------------------
// Relation phase: r_out[R,64] = r_feats @ W_R^T + b_R  (R=200, guarded tail)
// Tiny (13 tiles); weights stay in L2, no LDS staging needed.
// ---------------------------------------------------------------------------
__global__ void rel_transform_kernel(const float* __restrict__ r_feats,
                                     const float* __restrict__ W_R_w,
                                     const float* __restrict__ W_R_b,
                                     float* __restrict__ out,
                                     int R) {
  int wavesPerBlock = blockDim.x >> 5;
  int numTiles = (R + 15) >> 4;
  int tile = blockIdx.x * wavesPerBlock + (threadIdx.x >> 5);
  if (tile >= numTiles) return;  // wave-uniform

  int lane = threadIdx.x & 31;
  int lm   = lane & 15;
  int hs   = lane >> 4;
  int kb   = hs * 2;
  int mOff = hs * 8;
  int rowBase = tile * 16;

  int mA = rowBase + lm;
  bool aOk = (mA < R);
  const float* arow = r_feats + (size_t)(aOk ? mA : 0) * DD;

  const float* Wrow[4];
#pragma unroll
  for (int t = 0; t < 4; ++t) Wrow[t] = W_R_w + (size_t)(t * 16 + lm) * DD;

  v8f c[4];
#pragma unroll
  for (int t = 0; t < 4; ++t) c[t] = (v8f){0.f, 0.f, 0.f, 0.f, 0.f, 0.f, 0.f, 0.f};

#pragma unroll
  for (int k0 = 0; k0 < DD; k0 += 4) {
    v2f a = *(const v2f*)(arow + k0 + kb);
    if (!aOk) { a.x = 0.f; a.y = 0.f; }  // value select only; EXEC stays full
#pragma unroll
    for (int t = 0; t < 4; ++t) {
      v2f b = *(const v2f*)(Wrow[t] + k0 + kb);
      c[t] = wmma_f32x4(a, b, c[t]);
    }
  }

#pragma unroll
  for (int t = 0; t < 4; ++t) {
    int n = t * 16 + lm;
    float bias = W_R_b[n];
#pragma unroll
    for (int i = 0; i < 8; ++i) {
      int m = rowBase + i + mOff;
      if (m < R) out[(size_t)m * DD + n] = c[t][i] + bias;
    }
  }
}

// ---------------------------------------------------------------------------
extern "C" void kernel_launch(void* const* d_in, const int* in_sizes, int n_in,
                              void* d_out, int out_size, void* d_ws, size_t ws_size,
                              hipStream_t stream) {
  const float* n_feats  = (const float*)d_in[0];
  const float* r_feats  = (const float*)d_in[1];
  const float* W_I_w    = (const float*)d_in[2];
  const float* W_I_b    = (const float*)d_in[3];
  const float* W_O_w    = (const float*)d_in[4];
  const float* W_O_b    = (const float*)d_in[5];
  const float* W_R_w    = (const float*)d_in[6];
  const float* W_R_b    = (const float*)d_in[7];
  const int*   src      = (const int*)d_in[8];
  const int*   dst      = (const int*)d_in[9];
  const int*   etype    = (const int*)d_in[10];
  const int*   num_rels = (const int*)d_in[11];

  int N = in_sizes[0] / DD;   // 50000
  int R = in_sizes[1] / DD;   // 200
  int E = in_sizes[8];        // 1000000

  // Workspace layout: accI[N*64] | accO[N*64] | cntI[N] | cntO[N]  (~26 MB)
  float* accI = (float*)d_ws;
  float* accO = accI + (size_t)N * DD;
  float* cntI = accO + (size_t)N * DD;
  float* cntO = cntI + N;

  long long zcount = (long long)N * (2 * DD + 2);
  zero_kernel<<<1024, 256, 0, stream>>>(accI, zcount);

  // one wave32 per edge, 8 edges per 256-thread block
  int edgeBlocks = (E + 7) / 8;
  edge_scatter_kernel<<<edgeBlocks, 256, 0, stream>>>(
      n_feats, r_feats, src, dst, etype, num_rels,
      accI, accO, cntI, cntO, E);

  int nodeTiles = (N + 15) / 16;  // 3125 (N divisible by 16)
  node_transform_kernel<<<(nodeTiles + 3) / 4, 128, 0, stream>>>(
      accI, accO, cntI, cntO, W_I_w, W_I_b, W_O_w, W_O_b,
      (float*)d_out, nodeTiles);

  int relTiles = (R + 15) / 16;   // 13
  rel_transform_kernel<<<(relTiles + 3) / 4, 128, 0, stream>>>(
      r_feats, W_R_w, W_R_b, (float*)d_out + (size_t)N * DD, R);
}